// GIN_66864050864374
// MI455X (gfx1250) — compile-verified
//
#include <hip/hip_runtime.h>

typedef __attribute__((ext_vector_type(2))) float v2f;
typedef __attribute__((ext_vector_type(8))) float v8f;

#define N_NODES 10000
#define N_EDGES 640000
#define IN_C    128
#define HID     256

// ---------------------------------------------------------------- utilities
__global__ void zero_kernel(float* __restrict__ p, int n) {
    int i = blockIdx.x * blockDim.x + threadIdx.x;
    if (i < n) p[i] = 0.0f;
}

// agg[dst[e]][c] += x[src[e]][c]; one thread per (edge, 4-channel chunk)
__global__ void scatter_add_kernel(const float* __restrict__ x,
                                   const int* __restrict__ src,
                                   const int* __restrict__ dst,
                                   float* __restrict__ agg, int C) {
    int chunks = C >> 2;
    long long i = (long long)blockIdx.x * blockDim.x + threadIdx.x;
    long long total = (long long)N_EDGES * chunks;
    if (i >= total) return;
    int e  = (int)(i / chunks);
    int c4 = ((int)(i % chunks)) << 2;
    int s = src[e], d = dst[e];
    const float4 v = *(const float4*)(x + (long long)s * C + c4);
    float* o = agg + (long long)d * C + c4;
    atomicAdd(o + 0, v.x);
    atomicAdd(o + 1, v.y);
    atomicAdd(o + 2, v.z);
    atomicAdd(o + 3, v.w);
}

// WT[n*K + k] = W[k*N + n]  (makes WMMA B-fragment loads contiguous b64s)
__global__ void transpose_kernel(const float* __restrict__ W,
                                 float* __restrict__ WT, int K, int N) {
    int i = blockIdx.x * blockDim.x + threadIdx.x;
    if (i >= K * N) return;
    int k = i / N, n = i % N;
    WT[n * K + k] = W[i];
}

// ------------------------------------------------------------- WMMA GEMM
// Out[M,256] = maybe_relu( (A [+ A2]) @ W + bias ), W passed transposed as
// WT[256,K]. Each wave computes TWO 16x64 tiles (a 32x64 strip) so every
// B-fragment load feeds two v_wmma_f32_16x16x4_f32 ops. Block = 8 waves
// covering 64 rows x 256 cols. ADD_A/RELU are compile-time so the inner
// loop is pure loads + WMMA (no per-iteration branches).
template <int ADD_A, int RELU>
__global__ __launch_bounds__(256)
void gin_gemm_wmma(const float* __restrict__ A,
                   const float* __restrict__ A2,
                   const float* __restrict__ WT,
                   const float* __restrict__ bias,
                   float* __restrict__ Out,
                   int M, int K) {
    const int N    = 256;
    const int w    = threadIdx.x >> 5;   // wave id in block (wave32)
    const int lane = threadIdx.x & 31;
    const int half = lane >> 4;          // 0: K=k,k+1   1: K=k+2,k+3
    const int l15  = lane & 15;

    const int mpair = blockIdx.x * 2 + (w >> 2);  // pair of 16-row M-tiles
    const int m0 = mpair * 32;
    if (m0 >= M) return;                 // wave-uniform: EXEC stays all-ones
    const bool tile1_valid = (m0 + 16) < M;  // wave-uniform
    const int n0 = (w & 3) * 64;

    const int row0 = m0 + l15;
    int row1 = m0 + 16 + l15;
    if (row1 >= M) row1 = M - 1;         // clamp addresses only; EXEC untouched

    const float* A0r  = A  + (long long)row0 * K;
    const float* A1r  = A  + (long long)row1 * K;
    const float* A20r = A2 + (long long)row0 * K;
    const float* A21r = A2 + (long long)row1 * K;

    // prefetch the head of each B column strip (global_prefetch_b8)
#pragma unroll
    for (int t = 0; t < 4; ++t)
        __builtin_prefetch(WT + (long long)(n0 + 16 * t + l15) * K, 0, 0);

    v8f acc[2][4];
#pragma unroll
    for (int m = 0; m < 2; ++m)
#pragma unroll
        for (int t = 0; t < 4; ++t) acc[m][t] = (v8f){};

    for (int k = 0; k < K; k += 4) {
        const int kk = k + 2 * half;
        v2f a0 = *(const v2f*)(A0r + kk);      // A[row][kk], A[row][kk+1]
        v2f a1 = *(const v2f*)(A1r + kk);
        if (ADD_A) {                           // fused h = x + agg
            a0 = a0 + *(const v2f*)(A20r + kk);
            a1 = a1 + *(const v2f*)(A21r + kk);
        }
#pragma unroll
        for (int t = 0; t < 4; ++t) {
            const int col = n0 + 16 * t + l15;
            // B layout: v0 = B[kk][col], v1 = B[kk+1][col]; contiguous in WT
            v2f b = *(const v2f*)(WT + (long long)col * K + kk);
            acc[0][t] = __builtin_amdgcn_wmma_f32_16x16x4_f32(
                false, a0, false, b, (short)0, acc[0][t], false, false);
            acc[1][t] = __builtin_amdgcn_wmma_f32_16x16x4_f32(
                false, a1, false, b, (short)0, acc[1][t], false, false);
        }
    }

    // epilogue: bias + optional ReLU + store
#pragma unroll
    for (int m = 0; m < 2; ++m) {
        if (m == 1 && !tile1_valid) break;     // wave-uniform
#pragma unroll
        for (int t = 0; t < 4; ++t) {
            const int col = n0 + 16 * t + l15;
            const float bv = bias[col];
#pragma unroll
            for (int r = 0; r < 8; ++r) {
                float v = acc[m][t][r] + bv;
                if (RELU) v = fmaxf(v, 0.0f);
                const int rowOut = m0 + 16 * m + 8 * half + r;  // C/D layout
                Out[(long long)rowOut * N + col] = v;
            }
        }
    }
}

// ---------------------------------------------------------------- driver
extern "C" void kernel_launch(void* const* d_in, const int* in_sizes, int n_in,
                              void* d_out, int out_size, void* d_ws, size_t ws_size,
                              hipStream_t stream) {
    const float* x   = (const float*)d_in[0];
    const int*   ei  = (const int*)d_in[1];
    const int*   src = ei;             // edge_index[0]
    const int*   dst = ei + N_EDGES;   // edge_index[1]
    const float* W0a = (const float*)d_in[2];
    const float* b0a = (const float*)d_in[3];
    const float* W0b = (const float*)d_in[4];
    const float* b0b = (const float*)d_in[5];
    const float* W1a = (const float*)d_in[6];
    const float* b1a = (const float*)d_in[7];
    const float* W1b = (const float*)d_in[8];
    const float* b1b = (const float*)d_in[9];
    const float* W2a = (const float*)d_in[10];
    const float* b2a = (const float*)d_in[11];
    const float* W2b = (const float*)d_in[12];
    const float* b2b = (const float*)d_in[13];

    char* ws = (char*)d_ws;
    const size_t featBytes = (size_t)N_NODES * HID * sizeof(float); // 10.24 MB
    float* agg  = (float*)(ws);
    float* hmid = (float*)(ws + featBytes);
    float* hcur = (float*)(ws + 2 * featBytes);
    float* WT   = (float*)(ws + 3 * featBytes);   // 256 KB max

    const int nMTiles = N_NODES / 16;             // 625 (exact)
    const int nPairs  = (nMTiles + 1) / 2;        // 313
    const int gridM   = (nPairs + 1) / 2;         // 157 blocks of 8 waves

    auto zero = [&](float* p, int n) {
        zero_kernel<<<(n + 255) / 256, 256, 0, stream>>>(p, n);
    };
    auto scatter = [&](const float* h, int C) {
        long long total = (long long)N_EDGES * (C >> 2);
        scatter_add_kernel<<<(int)((total + 255) / 256), 256, 0, stream>>>(
            h, src, dst, agg, C);
    };
    auto gemm = [&](const float* A, const float* A2, int addA,
                    const float* W, const float* b, float* out, int K, int relu) {
        transpose_kernel<<<(K * 256 + 255) / 256, 256, 0, stream>>>(W, WT, K, 256);
        if (addA) {
            gin_gemm_wmma<1, 1><<<gridM, 256, 0, stream>>>(
                A, A2, WT, b, out, N_NODES, K);
        } else if (relu) {
            gin_gemm_wmma<0, 1><<<gridM, 256, 0, stream>>>(
                A, A, WT, b, out, N_NODES, K);
        } else {
            gin_gemm_wmma<0, 0><<<gridM, 256, 0, stream>>>(
                A, A, WT, b, out, N_NODES, K);
        }
    };

    // ----- layer 0 (IN_C=128 -> 256 -> 256), trailing ReLU
    zero(agg, N_NODES * IN_C);
    scatter(x, IN_C);
    gemm(x,    agg,     1, W0a, b0a, hmid, IN_C, 1);
    gemm(hmid, nullptr, 0, W0b, b0b, hcur, HID,  1);

    // ----- layer 1 (256 -> 256 -> 256), trailing ReLU
    zero(agg, N_NODES * HID);
    scatter(hcur, HID);
    gemm(hcur, agg,     1, W1a, b1a, hmid, HID, 1);
    gemm(hmid, nullptr, 0, W1b, b1b, hcur, HID, 1);

    // ----- layer 2 (256 -> 256 -> 256), no trailing ReLU
    zero(agg, N_NODES * HID);
    scatter(hcur, HID);
    gemm(hcur, agg,     1, W2a, b2a, hmid, HID, 1);
    gemm(hmid, nullptr, 0, W2b, b2b, (float*)d_out, HID, 0);
}